// SpanBert_2259152797757
// MI455X (gfx1250) — compile-verified
//
#include <hip/hip_runtime.h>
#include <hip/hip_bf16.h>

// ---------------------------------------------------------------------------
// SpanBERT mention + antecedent scoring, MI455X (gfx1250) version.
// scores = span_repr @ w_ment is rank-1 -> all-span scoring reduces to two
// 768-dots per span against precomputed fused vectors; the full span_repr
// GEMM chain runs only for the 4x128 selected spans. All GEMMs use
// V_WMMA_F32_16X16X4_F32 (full f32 matrix pipe). Inner loops are branch-free:
// K-padding via a zero-padded W_out copy, column handling via clamp+mask.
// ---------------------------------------------------------------------------

#define H 768
#define SEQ 2048
#define NB 4
#define NSPAN 4096
#define TOPK 128
#define WD 30
#define HID 150
#define CATW 1568   // 2H + WDIM (1566) padded to /16

typedef float v2f __attribute__((ext_vector_type(2)));
typedef float v8f __attribute__((ext_vector_type(8)));

__device__ __forceinline__ v8f v8f_zero() {
  v8f z = {0.f,0.f,0.f,0.f,0.f,0.f,0.f,0.f};
  return z;
}

// D = A(16x4) * B(4x16) + C, f32, wave32
__device__ __forceinline__ v8f wmma4(v2f a, v2f b, v8f c) {
  return __builtin_amdgcn_wmma_f32_16x16x4_f32(
      /*neg_a=*/false, a, /*neg_b=*/false, b,
      /*c_mod=*/(short)0, c, /*reuse_a=*/false, /*reuse_b=*/false);
}

// ---------------------------------------------------------------------------
__global__ void k_zero(float* __restrict__ p, int n) {
  int i = blockIdx.x * blockDim.x + threadIdx.x;
  if (i < n) p[i] = 0.f;
}

// Zero-padded copy of W_out: (1566 x 768) -> (1568 x 768), rows 1566/67 = 0.
__global__ void k_padW(const float* __restrict__ W_out, float* __restrict__ Wp) {
  int i = blockIdx.x * blockDim.x + threadIdx.x;
  if (i < CATW * H) Wp[i] = (i < (2 * H + WD) * H) ? W_out[i] : 0.f;
}

// ---------------------------------------------------------------------------
// Precompute fused score vectors:
//   a1 = W_out[0:H] @ w_ment ; a2 = W_out[H:2H] @ w_ment ; a3 = W_out[2H:] @ w_ment
//   s_start = W_start @ a1 ; s_end = W_end @ a2 ; wscore[w] = width_emb[w].a3
//   c0 = b_start.a1 + b_end.a2 + b_out.w_ment + b_ment
__global__ void k_prep(const float* __restrict__ W_start, const float* __restrict__ b_start,
                       const float* __restrict__ W_end,   const float* __restrict__ b_end,
                       const float* __restrict__ wemb,    const float* __restrict__ W_out,
                       const float* __restrict__ b_out,   const float* __restrict__ w_ment,
                       const float* __restrict__ b_ment,
                       float* __restrict__ sstart, float* __restrict__ send,
                       float* __restrict__ wscore, float* __restrict__ c0) {
  __shared__ float a1s[H], a2s[H], a3s[32];
  int h = threadIdx.x;                       // 768 threads
  float acc1 = 0.f, acc2 = 0.f;
  for (int d = 0; d < H; ++d) {
    float wm = w_ment[d];
    acc1 += W_out[(size_t)h * H + d] * wm;
    acc2 += W_out[(size_t)(H + h) * H + d] * wm;
  }
  a1s[h] = acc1;
  a2s[h] = acc2;
  if (h < WD) {
    float a3 = 0.f;
    for (int d = 0; d < H; ++d) a3 += W_out[(size_t)(2 * H + h) * H + d] * w_ment[d];
    a3s[h] = a3;
  }
  __syncthreads();
  float ss = 0.f, se = 0.f;
  for (int k = 0; k < H; ++k) {
    ss += W_start[(size_t)h * H + k] * a1s[k];
    se += W_end[(size_t)h * H + k] * a2s[k];
  }
  sstart[h] = ss;
  send[h]   = se;
  if (h < 11) {
    float w = 0.f;
    for (int c = 0; c < WD; ++c) w += wemb[h * WD + c] * a3s[c];
    wscore[h] = w;
  }
  if (h == 0) {
    float c = b_ment[0];
    for (int k = 0; k < H; ++k) c += b_start[k] * a1s[k] + b_end[k] * a2s[k];
    for (int d = 0; d < H; ++d) c += b_out[d] * w_ment[d];
    c0[0] = c;
  }
}

// ---------------------------------------------------------------------------
// One wave per span: score = seq[start].s_start + seq[end].s_end + wscore + c0
__global__ void k_scores(const float* __restrict__ seq, const int* __restrict__ starts,
                         const int* __restrict__ ends, const float* __restrict__ sstart,
                         const float* __restrict__ send, const float* __restrict__ wscore,
                         const float* __restrict__ c0, float* __restrict__ scores) {
  int id   = blockIdx.x * 8 + (threadIdx.x >> 5);   // 0..16383
  int lane = threadIdx.x & 31;
  int s = starts[id], e = ends[id];
  int b = id >> 12;
  const float4* rs  = (const float4*)(seq + ((size_t)b * SEQ + s) * H);
  const float4* re  = (const float4*)(seq + ((size_t)b * SEQ + e) * H);
  const float4* ws4 = (const float4*)sstart;
  const float4* we4 = (const float4*)send;
  float acc = 0.f;
#pragma unroll
  for (int t = 0; t < 6; ++t) {
    float4 x = rs[lane + 32 * t], w = ws4[lane + 32 * t];
    acc += x.x * w.x + x.y * w.y + x.z * w.z + x.w * w.w;
    float4 y = re[lane + 32 * t], v = we4[lane + 32 * t];
    acc += y.x * v.x + y.y * v.y + y.z * v.z + y.w * v.w;
  }
#pragma unroll
  for (int o = 16; o > 0; o >>= 1) acc += __shfl_down(acc, o);
  if (lane == 0) {
    int wd = e - s; wd = wd < 0 ? 0 : (wd > 10 ? 10 : wd);
    scores[id] = acc + wscore[wd] + c0[0];
  }
}

// ---------------------------------------------------------------------------
// Exact top-128 per batch: iterative argmax (descending, lowest index on ties
// -- matches jax.lax.top_k). One block per batch.
__global__ void k_topk(const float* __restrict__ scores, float* __restrict__ ment,
                       int* __restrict__ idx_out) {
  __shared__ float sv[NSPAN];
  __shared__ float rv[256];
  __shared__ int   ri[256];
  int b = blockIdx.x, tid = threadIdx.x;
  for (int j = tid; j < NSPAN; j += 256) sv[j] = scores[b * NSPAN + j];
  __syncthreads();
  for (int t = 0; t < TOPK; ++t) {
    float best = -3.4e38f; int bi = NSPAN;
    for (int j = tid; j < NSPAN; j += 256) {
      float v = sv[j];
      if (v > best || (v == best && j < bi)) { best = v; bi = j; }
    }
    rv[tid] = best; ri[tid] = bi;
    __syncthreads();
    for (int s = 128; s > 0; s >>= 1) {
      if (tid < s) {
        if (rv[tid + s] > rv[tid] || (rv[tid + s] == rv[tid] && ri[tid + s] < ri[tid])) {
          rv[tid] = rv[tid + s]; ri[tid] = ri[tid + s];
        }
      }
      __syncthreads();
    }
    if (tid == 0) {
      ment[b * TOPK + t]    = rv[0];
      idx_out[b * TOPK + t] = ri[0];
      sv[ri[0]] = -3.4e38f;
    }
    __syncthreads();
  }
}

// ---------------------------------------------------------------------------
// Gather selected spans: st/en rows, width emb into CAT tail, f_starts/f_ends.
__global__ void k_gather(const float* __restrict__ seq, const int* __restrict__ starts,
                         const int* __restrict__ ends, const int* __restrict__ idx,
                         const float* __restrict__ wemb, float* __restrict__ stk,
                         float* __restrict__ enk, float* __restrict__ cat,
                         float* __restrict__ fst, float* __restrict__ fen) {
  int id   = blockIdx.x * 8 + (threadIdx.x >> 5);   // 0..511
  int lane = threadIdx.x & 31;
  int b = id >> 7, k = id & 127;
  int m = idx[b * TOPK + k];
  int s = starts[b * NSPAN + m];
  int e = ends[b * NSPAN + m];
  const float4* rs = (const float4*)(seq + ((size_t)b * SEQ + s) * H);
  const float4* re = (const float4*)(seq + ((size_t)b * SEQ + e) * H);
  float4* ds = (float4*)(stk + (size_t)id * H);
  float4* de = (float4*)(enk + (size_t)id * H);
#pragma unroll
  for (int t = 0; t < 6; ++t) {
    ds[lane + 32 * t] = rs[lane + 32 * t];
    de[lane + 32 * t] = re[lane + 32 * t];
  }
  int w = e - s; w = w < 0 ? 0 : (w > 10 ? 10 : w);
  float* crow = cat + (size_t)id * CATW + 2 * H;
  crow[lane] = (lane < WD) ? wemb[w * WD + lane] : 0.f;   // cols 1566/1567 zeroed
  if (lane == 0) { fst[id] = (float)s; fen[id] = (float)e; }
}

// ---------------------------------------------------------------------------
// f32 WMMA GEMM: C[512 x Ncols] = A[512 x K4] * B + bias. K4 % 4 == 0, all
// loads unconditional (inputs pre-padded); out-of-N columns handled by
// clamp+mask (cndmask, no branches). Block tile 64x128: 8 waves of 32x32.
// MODE 0: B plain (ldb)    MODE 1: B = W1[k]+W1[1536+k]   MODE 2: W1[768+k]-W1[1536+k]
template <int MODE>
__global__ void k_gemm(const float* __restrict__ A, int lda,
                       const float* __restrict__ B, int ldb,
                       float* __restrict__ C, int ldc, const float* __restrict__ bias,
                       int Ncols, int K4) {
  int wave = threadIdx.x >> 5, lane = threadIdx.x & 31;
  int m0 = blockIdx.x * 64  + (wave >> 2) * 32;
  int n0 = blockIdx.y * 128 + (wave & 3) * 32;
  if (n0 >= Ncols) return;                   // whole wave right of N: skip
  int row = lane & 15;
  int khi = (lane >> 4) << 1;
  // clamped columns + 0/1 masks (branch-free OOB handling)
  int c0c = n0 + row,       c1c = n0 + 16 + row;
  int c0s = c0c < Ncols ? c0c : Ncols - 1;
  int c1s = c1c < Ncols ? c1c : Ncols - 1;
  float msk0 = c0c < Ncols ? 1.f : 0.f;
  float msk1 = c1c < Ncols ? 1.f : 0.f;
  v8f acc[2][2];
  acc[0][0] = v8f_zero(); acc[0][1] = v8f_zero();
  acc[1][0] = v8f_zero(); acc[1][1] = v8f_zero();
  const float* Ar0 = A + (size_t)(m0 + row) * lda;
  const float* Ar1 = A + (size_t)(m0 + 16 + row) * lda;
  const float* Bc0 = B + c0s;
  const float* Bc1 = B + c1s;
  for (int k0 = 0; k0 < K4; k0 += 4) {
    int ka = k0 + khi;
    v2f a0 = *(const v2f*)(Ar0 + ka);        // 8B-aligned pair load
    v2f a1 = *(const v2f*)(Ar1 + ka);
    v2f b0, b1;
    if (MODE == 0) {
      b0.x = Bc0[(size_t)ka * ldb];          b0.y = Bc0[(size_t)(ka + 1) * ldb];
      b1.x = Bc1[(size_t)ka * ldb];          b1.y = Bc1[(size_t)(ka + 1) * ldb];
    } else if (MODE == 1) {
      b0.x = Bc0[(size_t)ka * HID] + Bc0[(size_t)(1536 + ka) * HID];
      b0.y = Bc0[(size_t)(ka + 1) * HID] + Bc0[(size_t)(1537 + ka) * HID];
      b1.x = Bc1[(size_t)ka * HID] + Bc1[(size_t)(1536 + ka) * HID];
      b1.y = Bc1[(size_t)(ka + 1) * HID] + Bc1[(size_t)(1537 + ka) * HID];
    } else {
      b0.x = Bc0[(size_t)(768 + ka) * HID] - Bc0[(size_t)(1536 + ka) * HID];
      b0.y = Bc0[(size_t)(769 + ka) * HID] - Bc0[(size_t)(1537 + ka) * HID];
      b1.x = Bc1[(size_t)(768 + ka) * HID] - Bc1[(size_t)(1536 + ka) * HID];
      b1.y = Bc1[(size_t)(769 + ka) * HID] - Bc1[(size_t)(1537 + ka) * HID];
    }
    if (MODE != 0) { b0.x *= msk0; b0.y *= msk0; b1.x *= msk1; b1.y *= msk1; }
    acc[0][0] = wmma4(a0, b0, acc[0][0]);
    acc[0][1] = wmma4(a0, b1, acc[0][1]);
    acc[1][0] = wmma4(a1, b0, acc[1][0]);
    acc[1][1] = wmma4(a1, b1, acc[1][1]);
  }
  int ro = (lane >> 4) << 3;   // +8 rows for lanes 16..31
  int cc = lane & 15;
#pragma unroll
  for (int mi = 0; mi < 2; ++mi)
#pragma unroll
    for (int ni = 0; ni < 2; ++ni) {
      int c = n0 + ni * 16 + cc;
      if (c >= Ncols) continue;
      float bv = bias ? bias[c] : 0.f;
#pragma unroll
      for (int v = 0; v < 8; ++v) {
        int r = m0 + mi * 16 + ro + v;
        C[(size_t)r * ldc + c] = acc[mi][ni][v] + bv;
      }
    }
}

// ---------------------------------------------------------------------------
// Pairwise antecedent scores. Block = (i, b). Wave w owns j-tile [16w,16w+16)
// x all 150 d's (10 WMMA accumulators). pw[j,d] = sum_h r[b,j,h]*r[b,i,h]*Wd[h,d]
// via B[h,d] = r_i[h]*Wd[h,d] built on the fly. Waves above the diagonal exit
// early (tril k=-1). d-tiles 0..8 are full (c<=143<150); tile 9 uses
// clamp+mask. Epilogue: +u(i)+v(j)+b1, ReLU, LDS-staged per-row LayerNorm,
// dot with W2, store lower-triangular entry.
__global__ void k_pair(const float* __restrict__ repr, const float* __restrict__ Wd,
                       const float* __restrict__ u, const float* __restrict__ vrow,
                       const float* __restrict__ b1, const float* __restrict__ g,
                       const float* __restrict__ lb, const float* __restrict__ W2,
                       const float* __restrict__ b2, float* __restrict__ ant) {
  extern __shared__ float zbuf[];           // 8 waves * 16 * 160 floats = 80 KB
  int i = blockIdx.x, b = blockIdx.y;
  int wave = threadIdx.x >> 5, lane = threadIdx.x & 31;
  int j0 = wave * 16;
  if (j0 >= i) return;                      // no j < i in this tile
  const float* rb = repr + (size_t)(b * TOPK) * H;
  const float* ri = repr + (size_t)(b * TOPK + i) * H;
  int row = lane & 15;
  int khi = (lane >> 4) << 1;
  int c9  = 144 + row;
  int c9s = c9 < HID ? c9 : HID - 1;        // clamp last tile's column
  float msk9 = c9 < HID ? 1.f : 0.f;
  v8f acc[10];
#pragma unroll
  for (int t = 0; t < 10; ++t) acc[t] = v8f_zero();
  const float* Ar = rb + (size_t)(j0 + row) * H;
  for (int k0 = 0; k0 < H; k0 += 4) {
    int ka = k0 + khi;
    v2f a = *(const v2f*)(Ar + ka);
    v2f r = *(const v2f*)(ri + ka);
    const float* WA = Wd + (size_t)ka * HID;
    const float* WB = WA + HID;
#pragma unroll
    for (int t = 0; t < 9; ++t) {
      int c = t * 16 + row;
      v2f bb;
      bb.x = r.x * WA[c];
      bb.y = r.y * WB[c];
      acc[t] = wmma4(a, bb, acc[t]);
    }
    {
      v2f bb;
      bb.x = msk9 * r.x * WA[c9s];
      bb.y = msk9 * r.y * WB[c9s];
      acc[9] = wmma4(a, bb, acc[9]);
    }
  }
  // epilogue: z = pw + u_i + v_j + b1, ReLU, stage tile in LDS
  float* zt = zbuf + wave * 16 * 160;
  const float* urow = u + (size_t)(b * TOPK + i) * 160;
  const float* vb   = vrow + (size_t)(b * TOPK) * 160;
  int jo = (lane >> 4) << 3;
#pragma unroll
  for (int t = 0; t < 9; ++t) {
    int d = t * 16 + row;
    float ub = urow[d] + b1[d];
#pragma unroll
    for (int vv = 0; vv < 8; ++vv) {
      int joff = jo + vv;
      float val = acc[t][vv] + ub + vb[(size_t)(j0 + joff) * 160 + d];
      zt[joff * 160 + d] = fmaxf(val, 0.f);
    }
  }
  { // last d-tile, masked
    float ub = msk9 * (urow[c9s] + b1[c9s]);
#pragma unroll
    for (int vv = 0; vv < 8; ++vv) {
      int joff = jo + vv;
      float val = msk9 * (acc[9][vv] + vb[(size_t)(j0 + joff) * 160 + c9s]) + ub;
      zt[joff * 160 + c9] = fmaxf(val * msk9, 0.f);
    }
  }
  // per-row LayerNorm + readout (lanes 0..15, one j-row each)
  if (lane < 16) {
    int j = j0 + lane;
    if (j < i) {
      const float* hr = zt + lane * 160;
      float s = 0.f;
      for (int d = 0; d < HID; ++d) s += hr[d];
      float mu = s * (1.f / HID);
      float vs = 0.f, dv = 0.f, bs = 0.f;
      for (int d = 0; d < HID; ++d) {
        float hm = hr[d] - mu;
        vs += hm * hm;
        dv += hm * g[d] * W2[d];
        bs += lb[d] * W2[d];
      }
      float inv = rsqrtf(vs * (1.f / HID) + 1e-5f);
      ant[((size_t)(b * TOPK) + i) * TOPK + j] = inv * dv + bs + b2[0];
    }
  }
}

// ---------------------------------------------------------------------------
extern "C" void kernel_launch(void* const* d_in, const int* in_sizes, int n_in,
                              void* d_out, int out_size, void* d_ws, size_t ws_size,
                              hipStream_t stream) {
  const float* seq     = (const float*)d_in[0];
  const int*   starts  = (const int*)  d_in[1];
  const int*   ends    = (const int*)  d_in[2];
  const float* W_start = (const float*)d_in[3];
  const float* b_start = (const float*)d_in[4];
  const float* W_end   = (const float*)d_in[5];
  const float* b_end   = (const float*)d_in[6];
  const float* wemb    = (const float*)d_in[7];
  const float* W_out   = (const float*)d_in[8];
  const float* b_out   = (const float*)d_in[9];
  const float* w_ment  = (const float*)d_in[10];
  const float* b_ment  = (const float*)d_in[11];
  const float* W1      = (const float*)d_in[12];
  const float* b1      = (const float*)d_in[13];
  const float* ln_g    = (const float*)d_in[14];
  const float* ln_b    = (const float*)d_in[15];
  const float* W2      = (const float*)d_in[16];
  const float* b2      = (const float*)d_in[17];
  (void)in_sizes; (void)n_in; (void)ws_size;

  // workspace layout (floats); total ~13.5 MB
  float* wsf      = (float*)d_ws;
  float* w_sstart = wsf;                       // 768
  float* w_send   = wsf + 768;                 // 768
  float* w_wscore = wsf + 1536;                // 16
  float* w_c0     = wsf + 1552;                // 1 (pad to 1600)
  float* w_scores = wsf + 1600;                // 16384
  int*   w_idx    = (int*)(wsf + 17984);       // 512
  float* w_stk    = wsf + 18496;               // 512*768
  float* w_enk    = w_stk + 512 * H;           // 512*768
  float* w_cat    = w_enk + 512 * H;           // 512*1568
  float* w_repr   = w_cat + 512 * CATW;        // 512*768
  float* w_u      = w_repr + 512 * H;          // 512*160
  float* w_v      = w_u + 512 * 160;           // 512*160
  float* w_wpad   = w_v + 512 * 160;           // 1568*768 zero-padded W_out

  float* out    = (float*)d_out;               // 67072 floats total
  float* o_ment = out;                         // 4*128
  float* o_ant  = out + 512;                   // 4*128*128
  float* o_fst  = out + 66048;                 // 4*128
  float* o_fen  = out + 66560;                 // 4*128

  k_zero<<<(67072 + 255) / 256, 256, 0, stream>>>(out, 67072);
  k_padW<<<(CATW * H + 255) / 256, 256, 0, stream>>>(W_out, w_wpad);

  k_prep<<<1, H, 0, stream>>>(W_start, b_start, W_end, b_end, wemb, W_out, b_out,
                              w_ment, b_ment, w_sstart, w_send, w_wscore, w_c0);

  k_scores<<<2048, 256, 0, stream>>>(seq, starts, ends, w_sstart, w_send,
                                     w_wscore, w_c0, w_scores);

  k_topk<<<NB, 256, 0, stream>>>(w_scores, o_ment, w_idx);

  k_gather<<<64, 256, 0, stream>>>(seq, starts, ends, w_idx, wemb,
                                   w_stk, w_enk, w_cat, o_fst, o_fen);

  // start_e / end_e into CAT (cols 0..767 / 768..1535)
  k_gemm<0><<<dim3(8, 6), 256, 0, stream>>>(w_stk, H, W_start, H,
                                            w_cat, CATW, b_start, H, H);
  k_gemm<0><<<dim3(8, 6), 256, 0, stream>>>(w_enk, H, W_end, H,
                                            w_cat + H, CATW, b_end, H, H);
  // span_repr = CAT @ W_out_padded + b_out   (K = 1568, zero-padded rows)
  k_gemm<0><<<dim3(8, 6), 256, 0, stream>>>(w_cat, CATW, w_wpad, H,
                                            w_repr, H, b_out, H, CATW);
  // u = repr @ (Wa+Wc) ; v = repr @ (Wb-Wc)   (N = 150, ldc = 160)
  k_gemm<1><<<dim3(8, 2), 256, 0, stream>>>(w_repr, H, W1, HID,
                                            w_u, 160, nullptr, HID, H);
  k_gemm<2><<<dim3(8, 2), 256, 0, stream>>>(w_repr, H, W1, HID,
                                            w_v, 160, nullptr, HID, H);

  const float* Wd = W1 + (size_t)(3 * H) * HID;
  k_pair<<<dim3(TOPK, NB), 256, 8 * 16 * 160 * sizeof(float), stream>>>(
      w_repr, Wd, w_u, w_v, b1, ln_g, ln_b, W2, b2, o_ant);
}